// Matern52_59794534695227
// MI455X (gfx1250) — compile-verified
//
#include <hip/hip_runtime.h>
#include <math.h>

typedef __attribute__((ext_vector_type(2))) float v2f;
typedef __attribute__((ext_vector_type(8))) float v8f;
typedef __attribute__((ext_vector_type(4))) int   v4i;

#define DIMS       64
#define LDS_STRIDE 68   // 64 + 4 pad floats: bank-conflict-free column access
#define BM         32   // block tile rows (2 waves of 16)
#define BN         64   // block tile cols (4 waves of 16)

#if defined(__has_builtin)
#  if __has_builtin(__builtin_amdgcn_global_load_async_to_lds_b128)
#    define HAVE_ASYNC_LDS 1
#  endif
#endif
#ifndef HAVE_ASYNC_LDS
#  define HAVE_ASYNC_LDS 0
#endif

typedef __attribute__((address_space(1))) v4i* gvec4_t;   // global int4*
typedef __attribute__((address_space(3))) v4i* lvec4_t;   // LDS int4*

static __device__ __forceinline__ void wait_async_all() {
#if defined(__has_builtin)
#  if __has_builtin(__builtin_amdgcn_s_wait_asynccnt)
    __builtin_amdgcn_s_wait_asynccnt(0);
#  else
    asm volatile("s_wait_asynccnt 0x0" ::: "memory");
#  endif
#else
    asm volatile("s_wait_asynccnt 0x0" ::: "memory");
#endif
}

__global__ __launch_bounds__(256)
void matern52_wmma_f32(const float* __restrict__ X1,
                       const float* __restrict__ X2,
                       const float* __restrict__ c,
                       const float* __restrict__ l,
                       float* __restrict__ out,
                       int N1, int N2)
{
    __shared__ __align__(16) float rl2[DIMS];               // (1/l)^2
    __shared__ __align__(16) float sq1s[BM];
    __shared__ __align__(16) float sq2s[BN];
    __shared__ __align__(16) float X1s[BM * LDS_STRIDE];    // raw X1 tile
    __shared__ __align__(16) float X2s[BN * LDS_STRIDE];    // raw X2 tile

    const int t  = threadIdx.x;
    const int m0 = blockIdx.y * BM;
    const int n0 = blockIdx.x * BN;

    if (t < DIMS) {
        float r = 1.0f / l[t];
        rl2[t] = r * r;
    }

#if HAVE_ASYNC_LDS
    // Async DMA raw tiles straight into LDS (16 B per lane per op, ASYNCcnt).
    #pragma unroll
    for (int i = 0; i < (BM * DIMS) / (256 * 4); ++i) {
        int idx = (t + i * 256) * 4;
        int r = idx >> 6, k = idx & 63;
        __builtin_amdgcn_global_load_async_to_lds_b128(
            (gvec4_t)(X1 + (size_t)(m0 + r) * DIMS + k),
            (lvec4_t)(&X1s[r * LDS_STRIDE + k]), 0, 0);
    }
    #pragma unroll
    for (int i = 0; i < (BN * DIMS) / (256 * 4); ++i) {
        int idx = (t + i * 256) * 4;
        int r = idx >> 6, k = idx & 63;
        __builtin_amdgcn_global_load_async_to_lds_b128(
            (gvec4_t)(X2 + (size_t)(n0 + r) * DIMS + k),
            (lvec4_t)(&X2s[r * LDS_STRIDE + k]), 0, 0);
    }
    wait_async_all();
#else
    // Synchronous fallback staging (raw tiles).
    #pragma unroll
    for (int i = 0; i < (BM * DIMS) / 256; ++i) {
        int idx = t + i * 256;
        int r = idx >> 6, k = idx & 63;
        X1s[r * LDS_STRIDE + k] = X1[(size_t)(m0 + r) * DIMS + k];
    }
    #pragma unroll
    for (int i = 0; i < (BN * DIMS) / 256; ++i) {
        int idx = t + i * 256;
        int r = idx >> 6, k = idx & 63;
        X2s[r * LDS_STRIDE + k] = X2[(size_t)(n0 + r) * DIMS + k];
    }
#endif
    __syncthreads();

    // Cooperative scaled row norms: sq = sum_k X[m][k]^2 * rl2[k]
    if (t < BM + BN) {
        const float* row = (t < BM) ? &X1s[t * LDS_STRIDE]
                                    : &X2s[(t - BM) * LDS_STRIDE];
        float s = 0.0f;
        #pragma unroll
        for (int k = 0; k < DIMS; ++k) s = fmaf(row[k] * row[k], rl2[k], s);
        if (t < BM) sq1s[t] = s;
        else        sq2s[t - BM] = s;
    }
    __syncthreads();

    // wave -> 16x16 sub-tile: waves arranged 2 (M) x 4 (N)
    const int lane = t & 31;
    const int wave = t >> 5;
    const int wm   = wave >> 2;   // 0..1
    const int wn   = wave & 3;    // 0..3
    const int lr   = lane & 15;
    const int hi   = lane >> 4;

    // f32 16x16x4 fragments:
    //   A: lane l, vgpr j -> A[M = l%16][K = k0 + 2*(l/16) + j]   (scaled by rl2)
    //   B: lane l, vgpr j -> B[K][N = l%16] = X2_raw[l%16][K]
    const float* aRow = &X1s[(wm * 16 + lr) * LDS_STRIDE];
    const float* bRow = &X2s[(wn * 16 + lr) * LDS_STRIDE];

    v8f acc = {};
    #pragma unroll
    for (int k0 = 0; k0 < DIMS; k0 += 4) {
        const int ko = k0 + 2 * hi;
        v2f a = *(const v2f*)(aRow + ko);
        v2f s = *(const v2f*)(rl2 + ko);     // broadcast within half-wave
        v2f b = *(const v2f*)(bRow + ko);
        a = a * s;                           // v_pk_mul_f32
        acc = __builtin_amdgcn_wmma_f32_16x16x4_f32(
            /*neg_a=*/false, a, /*neg_b=*/false, b,
            /*c_mod=*/(short)0, acc, /*reuse_a=*/false, /*reuse_b=*/false);
    }

    // Epilogue: Matern 5/2 transform
    const float c0 = c[0];
    const float c2 = c0 * c0;
    const float sq2v = sq2s[wn * 16 + lr];
    const float SQRT5 = 2.23606797749978969f;
    const float FIVE_THIRDS = 1.66666666666666667f;

    const int colg = n0 + wn * 16 + lr;
    #pragma unroll
    for (int v = 0; v < 8; ++v) {
        // C/D layout: lane l, vgpr v -> (M = v + 8*(l/16), N = l%16)
        int mloc = wm * 16 + v + 8 * hi;
        float d2 = sq1s[mloc] + sq2v - 2.0f * acc[v];
        d2 = fmaxf(d2, 0.0f);
        float r = sqrtf(d2);
        float val = c2 * (1.0f + SQRT5 * r + FIVE_THIRDS * d2) * __expf(-SQRT5 * r);
        out[(size_t)(m0 + mloc) * N2 + colg] = val;
    }
}

extern "C" void kernel_launch(void* const* d_in, const int* in_sizes, int n_in,
                              void* d_out, int out_size, void* d_ws, size_t ws_size,
                              hipStream_t stream) {
    const float* X1 = (const float*)d_in[0];
    const float* X2 = (const float*)d_in[1];
    const float* c  = (const float*)d_in[2];
    const float* l  = (const float*)d_in[3];
    float* out = (float*)d_out;

    const int N1 = in_sizes[0] / DIMS;   // 8192
    const int N2 = in_sizes[1] / DIMS;   // 8192

    dim3 grid((unsigned)(N2 / BN), (unsigned)(N1 / BM));
    matern52_wmma_f32<<<grid, 256, 0, stream>>>(X1, X2, c, l, out, N1, N2);
}